// MaskEstimatorGSS_67774583931373
// MI455X (gfx1250) — compile-verified
//
#include <hip/hip_runtime.h>
#include <math.h>

// ---------------- problem constants (from reference setup_inputs) ----------
#define BATCH 2
#define NCH   8          // num_inputs C
#define NF    513
#define NT    1000
#define NM    4          // num_outputs
#define NBMF  (BATCH*NM*NF)   // 4104
#define EPSV  1e-8f

typedef __attribute__((ext_vector_type(2))) float v2f;
typedef __attribute__((ext_vector_type(8))) float v8f;

__device__ __forceinline__ float fast_rcp(float x) {
  return __builtin_amdgcn_rcpf(x);   // v_rcp_f32
}

// ---------------------------------------------------------------------------
// K0: build z (directional stats, packed as 16 real rows [Re0..7, Im0..7] per
// (b,f), row-major over t), initial gamma, and initial scale
//   scl = gamma/(zh+EPS) with zh==1.
// ---------------------------------------------------------------------------
__global__ void k_setup(const float* __restrict__ xr, const float* __restrict__ xi,
                        const float* __restrict__ act,
                        float* __restrict__ z, float* __restrict__ gamma,
                        float* __restrict__ scl) {
  int idx = blockIdx.x * blockDim.x + threadIdx.x;
  if (idx >= BATCH * NF * NT) return;
  int t = idx % NT;
  int f = (idx / NT) % NF;
  int b = idx / (NT * NF);

  float re[NCH], im[NCH];
  float ss = 0.f;
  for (int c = 0; c < NCH; ++c) {
    re[c] = xr[((b * NCH + c) * NF + f) * NT + t];
    im[c] = xi[((b * NCH + c) * NF + f) * NT + t];
    ss += re[c] * re[c] + im[c] * im[c];
  }
  float inv = fast_rcp(sqrtf(ss) + EPSV);
  int zb = ((b * NF + f) * 16) * NT + t;
  for (int c = 0; c < NCH; ++c) {
    z[zb + c * NT]       = re[c] * inv;
    z[zb + (8 + c) * NT] = im[c] * inv;
  }

  // initial masks from activity: clip, normalize over M
  float g[NM], s = 0.f;
  for (int m = 0; m < NM; ++m) {
    float a = act[(b * NM + m) * NT + t];
    g[m] = a > EPSV ? a : EPSV;
    s += g[m];
  }
  float invs = fast_rcp(s);
  const float rz1 = 1.0f / (1.0f + EPSV);   // zh initialized to 1
  for (int m = 0; m < NM; ++m) {
    int gi = ((b * NM + m) * NF + f) * NT + t;
    float gm = g[m] * invs;
    gamma[gi] = gm;
    scl[gi] = gm * rz1;
  }
}

// ---------------------------------------------------------------------------
// K1: denom = sum_t gamma, alpha = mean_t gamma, per (b,m,f). One block per bmf.
// ---------------------------------------------------------------------------
__global__ void k_stats(const float* __restrict__ gamma,
                        float* __restrict__ denom, float* __restrict__ alpha) {
  __shared__ float red[256];
  int bmf = blockIdx.x;
  const float* g = gamma + (long)bmf * NT;
  float s = 0.f;
  for (int t = threadIdx.x; t < NT; t += blockDim.x) s += g[t];
  red[threadIdx.x] = s;
  __syncthreads();
  for (int off = 128; off > 0; off >>= 1) {
    if (threadIdx.x < off) red[threadIdx.x] += red[threadIdx.x + off];
    __syncthreads();
  }
  if (threadIdx.x == 0) {
    denom[bmf] = red[0];
    alpha[bmf] = red[0] * (1.0f / (float)NT);
  }
}

// ---------------------------------------------------------------------------
// K2: weighted covariance via WMMA f32 16x16x4.
// One wave32 per (b,m,f): G(16x16) = sum_t scl_t * v_t v_t^T
// A lane layout (16x4 f32): lanes 0-15 row M=l hold K=0/1 in v0/v1,
//                           lanes 16-31 row M=l hold K=2/3.
// B lane layout (4x16):     v0: K=(g?2:0), N=l ; v1: K=(g?3:1), N=l.
// ---------------------------------------------------------------------------
__global__ void k_cov(const float* __restrict__ z, const float* __restrict__ scl,
                      float* __restrict__ G) {
  int wave = threadIdx.x >> 5;
  int lane = threadIdx.x & 31;
  int bmf = blockIdx.x * (blockDim.x >> 5) + wave;
  if (bmf >= NBMF) return;
  int f = bmf % NF;
  int b = bmf / (NF * NM);
  int bf = b * NF + f;

  int g = lane >> 4;       // K half: 0 -> K{0,1}, 1 -> K{2,3}
  int l = lane & 15;       // row (A) / col (B)

  const float* zrow = z + ((long)(bf * 16 + l)) * NT;
  const float* wrow = scl + (long)bmf * NT;

  v8f acc = {};
  for (int t0 = 0; t0 < NT; t0 += 4) {
    int tg = t0 + 2 * g;
    v2f v = *(const v2f*)(zrow + tg);
    v2f w = *(const v2f*)(wrow + tg);
    v2f a;
    a.x = v.x * w.x;
    a.y = v.y * w.y;
    acc = __builtin_amdgcn_wmma_f32_16x16x4_f32(false, a, false, v,
                                                (short)0, acc, false, false);
  }
  // D layout: vgpr r -> row i = 8*g + r, col j = l
  float* out = G + (long)bmf * 256;
  for (int r = 0; r < 8; ++r) out[(8 * g + r) * 16 + l] = acc[r];
}

// ---------------------------------------------------------------------------
// K3: per-thread 8x8 complex-Hermitian Jacobi eig, build whitening matrix
// W16 = real 16x16 rep of diag(1/sqrt(L)) Q^H, and log|det| per (b,m,f).
// Matrices live in LDS: 64 threads * (128+128) floats = 64 KB.
// ---------------------------------------------------------------------------
#define AR(i,j) sA[tid][((i)*8+(j))*2]
#define AI(i,j) sA[tid][((i)*8+(j))*2+1]
#define QR(i,j) sQ[tid][((i)*8+(j))*2]
#define QI(i,j) sQ[tid][((i)*8+(j))*2+1]

__global__ void k_eig(const float* __restrict__ G, const float* __restrict__ denom,
                      float* __restrict__ W16, float* __restrict__ logdet) {
  __shared__ float sA[64][128];
  __shared__ float sQ[64][128];
  int tid = threadIdx.x;
  int bmf = blockIdx.x * 64 + tid;
  if (bmf >= NBMF) return;

  const float* g = G + (long)bmf * 256;
  float sc = (float)NCH * fast_rcp(denom[bmf] + EPSV);

  // Hermitian BM from real 16x16 G (symmetrize -> hermitize)
  for (int i = 0; i < 8; ++i)
    for (int j = 0; j < 8; ++j) {
      float re_ij = g[i * 16 + j] + g[(8 + i) * 16 + (8 + j)];
      float im_ij = g[(8 + i) * 16 + j] - g[i * 16 + (8 + j)];
      float re_ji = g[j * 16 + i] + g[(8 + j) * 16 + (8 + i)];
      float im_ji = g[(8 + j) * 16 + i] - g[j * 16 + (8 + i)];
      AR(i, j) = 0.5f * sc * (re_ij + re_ji);
      AI(i, j) = 0.5f * sc * (im_ij - im_ji);
      QR(i, j) = (i == j) ? 1.f : 0.f;
      QI(i, j) = 0.f;
    }

  // cyclic Jacobi sweeps
  for (int sweep = 0; sweep < 12; ++sweep) {
    for (int p = 0; p < 7; ++p)
      for (int q = p + 1; q < 8; ++q) {
        float ar = AR(p, q), ai = AI(p, q);
        float mag2 = ar * ar + ai * ai;
        if (mag2 < 1e-26f) continue;
        float mag = sqrtf(mag2);
        float rm = fast_rcp(mag);
        float gr = ar * rm, gi = ai * rm;   // phase of a_pq
        float tau = (AR(q, q) - AR(p, p)) * 0.5f * rm;
        float tt = (tau >= 0.f ? 1.f : -1.f) * fast_rcp(fabsf(tau) + sqrtf(1.f + tau * tau));
        float c = rsqrtf(1.f + tt * tt);
        float s = tt * c;
        // columns: A <- A*U, Q <- Q*U
        for (int k = 0; k < 8; ++k) {
          float pr = AR(k, p), pi = AI(k, p);
          float qr = AR(k, q), qi = AI(k, q);
          float cr = gr * qr + gi * qi, ci = gr * qi - gi * qr;  // conj(g)*a_kq
          AR(k, p) = c * pr - s * cr;  AI(k, p) = c * pi - s * ci;
          float dr = gr * pr - gi * pi, di = gr * pi + gi * pr;  // g*a_kp
          AR(k, q) = s * dr + c * qr;  AI(k, q) = s * di + c * qi;

          pr = QR(k, p); pi = QI(k, p);
          qr = QR(k, q); qi = QI(k, q);
          cr = gr * qr + gi * qi; ci = gr * qi - gi * qr;
          QR(k, p) = c * pr - s * cr;  QI(k, p) = c * pi - s * ci;
          dr = gr * pr - gi * pi; di = gr * pi + gi * pr;
          QR(k, q) = s * dr + c * qr;  QI(k, q) = s * di + c * qi;
        }
        // rows: A <- U^H * A
        for (int k = 0; k < 8; ++k) {
          float pr = AR(p, k), pi = AI(p, k);
          float qr = AR(q, k), qi = AI(q, k);
          float cr = gr * qr - gi * qi, ci = gr * qi + gi * qr;  // g*a_qk
          AR(p, k) = c * pr - s * cr;  AI(p, k) = c * pi - s * ci;
          float dr = gr * pr + gi * pi, di = gr * pi - gi * pr;  // conj(g)*a_pk
          AR(q, k) = s * dr + c * qr;  AI(q, k) = s * di + c * qi;
        }
      }
  }

  // eigenvalues -> clipped, normalized, whitening weights
  float L[8], Lmax = 0.f;
  for (int i = 0; i < 8; ++i) {
    float v = AR(i, i);
    L[i] = v > EPSV ? v : EPSV;
    Lmax = fmaxf(Lmax, L[i]);
  }
  float ld = 0.f, w[8];
  float invmax = fast_rcp(Lmax + EPSV);
  for (int i = 0; i < 8; ++i) {
    L[i] = L[i] * invmax + EPSV;
    ld += __logf(L[i]);
    w[i] = rsqrtf(L[i]);
  }
  logdet[bmf] = ld;

  // M = diag(w) Q^H ; real 16x16 block form [[Re, -Im],[Im, Re]]
  float* o = W16 + (long)bmf * 256;
  for (int j = 0; j < 8; ++j)
    for (int k = 0; k < 8; ++k) {
      float mre = w[j] * QR(k, j);
      float mim = -w[j] * QI(k, j);   // conj(Q[k][j])
      o[j * 16 + k]             = mre;
      o[j * 16 + (k + 8)]       = -mim;
      o[(j + 8) * 16 + k]       = mim;
      o[(j + 8) * 16 + (k + 8)] = mre;
    }
}

// ---------------------------------------------------------------------------
// K4: projection + zh update via WMMA. One wave32 per (b,m,f).
// Y(16 x 16t-tile) = W16(16x16) * V(16 x tile)  via 4 chained 16x16x4 WMMAs.
// zh[t] = sum_i Y[i][t]^2 + EPS  (cross half-wave reduce via shfl_xor 16).
// Writes into the shared zh/scale buffer (overwrites previous scale).
// ---------------------------------------------------------------------------
__global__ void k_proj(const float* __restrict__ z, const float* __restrict__ W16,
                       float* __restrict__ zh) {
  int wave = threadIdx.x >> 5;
  int lane = threadIdx.x & 31;
  int bmf = blockIdx.x * (blockDim.x >> 5) + wave;
  if (bmf >= NBMF) return;
  int f = bmf % NF;
  int b = bmf / (NF * NM);
  int bf = b * NF + f;

  int g = lane >> 4;
  int l = lane & 15;

  // preload A chunks: A_kc[i=l][k0+2g .. k0+2g+1]
  const float* wm = W16 + (long)bmf * 256 + l * 16;
  v2f A[4];
  for (int kc = 0; kc < 4; ++kc)
    A[kc] = *(const v2f*)(wm + 4 * kc + 2 * g);

  const float* zb = z + (long)(bf * 16) * NT;
  float* out = zh + (long)bmf * NT;

  int ntile = (NT + 15) / 16;
  for (int tile = 0; tile < ntile; ++tile) {
    int tcol = tile * 16 + l;
    int tc = tcol < NT ? tcol : NT - 1;
    v8f acc = {};
    for (int kc = 0; kc < 4; ++kc) {
      int k0 = 4 * kc + 2 * g;
      v2f bm;
      bm.x = zb[(long)k0 * NT + tc];
      bm.y = zb[(long)(k0 + 1) * NT + tc];
      acc = __builtin_amdgcn_wmma_f32_16x16x4_f32(false, A[kc], false, bm,
                                                  (short)0, acc, false, false);
    }
    float part = 0.f;
    for (int r = 0; r < 8; ++r) part += acc[r] * acc[r];
    part += __shfl_xor(part, 16, 32);
    if (g == 0 && tcol < NT) out[tcol] = part + EPSV;
  }
}

// ---------------------------------------------------------------------------
// K5: mask update per (b,f,t): softmax over M of log_pdf, weighted by
// alpha and activity. Also fuses next iteration's scale = gamma/(zh+EPS),
// overwriting zh in place (read-then-write per element).
// ---------------------------------------------------------------------------
__global__ void k_masks(float* __restrict__ zh_scl, const float* __restrict__ logdet,
                        const float* __restrict__ alpha, const float* __restrict__ act,
                        float* __restrict__ gamma_out) {
  int idx = blockIdx.x * blockDim.x + threadIdx.x;
  if (idx >= BATCH * NF * NT) return;
  int t = idx % NT;
  int f = (idx / NT) % NF;
  int b = idx / (NT * NF);

  float zv[NM], lp[NM], mx = -3.4e38f;
  for (int m = 0; m < NM; ++m) {
    int bmf = (b * NM + m) * NF + f;
    zv[m] = zh_scl[(long)bmf * NT + t];
    lp[m] = -(float)NCH * __logf(zv[m]) - logdet[bmf];
    mx = fmaxf(mx, lp[m]);
  }
  float gv[NM], s = 0.f;
  for (int m = 0; m < NM; ++m) {
    int bmf = (b * NM + m) * NF + f;
    float e = __expf(lp[m] - mx) * alpha[bmf] * act[(b * NM + m) * NT + t];
    gv[m] = e;
    s += e;
  }
  float inv = fast_rcp(s + EPSV);
  for (int m = 0; m < NM; ++m) {
    long gi = ((long)((b * NM + m) * NF + f)) * NT + t;
    float gm = gv[m] * inv;
    gamma_out[gi] = gm;
    zh_scl[gi] = gm * fast_rcp(zv[m] + EPSV);   // scale for next iteration
  }
}

// ---------------------------------------------------------------------------
extern "C" void kernel_launch(void* const* d_in, const int* in_sizes, int n_in,
                              void* d_out, int out_size, void* d_ws, size_t ws_size,
                              hipStream_t stream) {
  const float* xr  = (const float*)d_in[0];
  const float* xi  = (const float*)d_in[1];
  const float* act = (const float*)d_in[2];
  float* out = (float*)d_out;

  // workspace carve-up (floats)
  float* ws = (float*)d_ws;
  size_t off = 0;
  float* z      = ws + off; off += (size_t)BATCH * NF * 16 * NT;  // 16,416,000
  float* gamma  = ws + off; off += (size_t)NBMF * NT;             //  4,104,000
  float* zh     = ws + off; off += (size_t)NBMF * NT;             //  4,104,000 (zh / scale)
  float* G      = ws + off; off += (size_t)NBMF * 256;            //  1,050,624
  float* W16    = ws + off; off += (size_t)NBMF * 256;            //  1,050,624
  float* denom  = ws + off; off += NBMF;
  float* alpha  = ws + off; off += NBMF;
  float* logdet = ws + off; off += NBMF;

  int nbft = BATCH * NF * NT;
  int blk = 256;

  k_setup<<<(nbft + blk - 1) / blk, blk, 0, stream>>>(xr, xi, act, z, gamma, zh);

  for (int it = 0; it < 3; ++it) {
    k_stats<<<NBMF, 256, 0, stream>>>(gamma, denom, alpha);
    k_cov<<<(NBMF + 7) / 8, 256, 0, stream>>>(z, zh /*scale*/, G);
    k_eig<<<(NBMF + 63) / 64, 64, 0, stream>>>(G, denom, W16, logdet);
    k_proj<<<(NBMF + 7) / 8, 256, 0, stream>>>(z, W16, zh);
    float* gdst = (it == 2) ? out : gamma;
    k_masks<<<(nbft + blk - 1) / blk, blk, 0, stream>>>(zh, logdet, alpha, act, gdst);
  }
}